// _GSPostProcessor_79534204387746
// MI455X (gfx1250) — compile-verified
//
#include <hip/hip_runtime.h>
#include <stdint.h>

// ---------------------------------------------------------------------------
// GSPostProcessor: relu -> per-row exact top-K (K=20) sparsify -> +I -> sym-norm
// B=16, N=2048 -> 32768 independent rows of 2048 fp32.
// Roofline: 512MB in + 256MB out ~ 768MB @ 23.3TB/s ~= 33us. No matmul
// structure -> WMMA not applicable; optimize data movement:
//  - async global->LDS row staging (ASYNCcnt path) so A is read once
//  - NT b128 streaming stores for the mostly-zero dense output
//  - exact bitwise top-K (u32 patterns of non-negative floats are ordered)
// Workspace use: 32768*20*(4+4) + 32768*4 bytes ~= 5.4 MB in d_ws.
// ---------------------------------------------------------------------------

#define N_ 2048
#define B_ 16
#define K_ 20
#define NROWS (B_ * N_)
#define WAVES 8
#define CHUNKS 16 // 16 chunks * 32 lanes * 4 floats = 2048 cols
#define ELEMS 64  // per-lane elements

typedef float v4f __attribute__((ext_vector_type(4)));
typedef int v4i __attribute__((ext_vector_type(4)));

#if defined(__AMDGCN__) &&                                                    \
    __has_builtin(__builtin_amdgcn_global_load_async_to_lds_b128) &&          \
    __has_builtin(__builtin_amdgcn_s_wait_asynccnt)
#define USE_ASYNC_LDS 1
typedef __attribute__((address_space(1))) v4i GlobalV4i; // global (AS1) int4
typedef __attribute__((address_space(3))) v4i LdsV4i;    // LDS (AS3) int4
#else
#define USE_ASYNC_LDS 0
#endif

__device__ __forceinline__ int wredAdd(int v) {
#pragma unroll
  for (int m = 16; m; m >>= 1) v += __shfl_xor(v, m, 32);
  return __builtin_amdgcn_readfirstlane(v);
}
__device__ __forceinline__ unsigned wredMaxU(unsigned v) {
#pragma unroll
  for (int m = 16; m; m >>= 1) {
    unsigned o = (unsigned)__shfl_xor((int)v, m, 32);
    v = v > o ? v : o;
  }
  return (unsigned)__builtin_amdgcn_readfirstlane((int)v);
}
__device__ __forceinline__ int wredMinI(int v) {
#pragma unroll
  for (int m = 16; m; m >>= 1) {
    int o = __shfl_xor(v, m, 32);
    v = v < o ? v : o;
  }
  return __builtin_amdgcn_readfirstlane(v);
}
__device__ __forceinline__ float wredAddF(float v) {
#pragma unroll
  for (int m = 16; m; m >>= 1) v += __shfl_xor(v, m, 32);
  return v;
}

// Pass 1: one wave32 per row. Streams zeros into the output row, stages the
// A row in LDS (async), finds the exact K-th largest key by binary search on
// the u32 bit patterns, emits compact (col, relu) pairs + d_inv to workspace.
__global__ __launch_bounds__(WAVES * 32) void gsp_topk_rows(
    const float* __restrict__ A, const float* __restrict__ dope,
    float* __restrict__ out, int* __restrict__ wcols,
    float* __restrict__ wvals, float* __restrict__ wdinv) {
  __shared__ float sA[WAVES][N_];
  __shared__ unsigned slot[WAVES];

  const int lane = (int)threadIdx.x & 31;
  const int w = (int)threadIdx.x >> 5;
  const int row = (int)blockIdx.x * WAVES + w;
  const size_t rbase = (size_t)row * N_;
  const float* Arow = A + rbase;
  const float* Drow = dope + rbase;
  float* Orow = out + rbase;

  // Stream zeros into this row of the (mostly sparse) dense output.
  {
    v4f z = {0.f, 0.f, 0.f, 0.f};
#pragma unroll
    for (int c = 0; c < CHUNKS; ++c)
      __builtin_nontemporal_store(z, (v4f*)(Orow + c * 128 + lane * 4));
  }
  if (lane == 0) slot[w] = 0u; // per-wave LDS ops stay in order within a wave

  unsigned key[ELEMS];
  unsigned kmax = 0u;

#if USE_ASYNC_LDS
  // Async copy the whole A row into this wave's LDS slab (16B per lane per op).
#pragma unroll
  for (int c = 0; c < CHUNKS; ++c) {
    __builtin_amdgcn_global_load_async_to_lds_b128(
        (GlobalV4i*)(Arow + c * 128 + lane * 4),
        (LdsV4i*)(&sA[w][c * 128 + lane * 4]), 0, 0);
  }
  __builtin_amdgcn_s_wait_asynccnt(0);
#pragma unroll
  for (int c = 0; c < CHUNKS; ++c) {
    v4f a4 = *(v4f*)(&sA[w][c * 128 + lane * 4]); // ds_load_b128
    v4f d4 = *(const v4f*)(Drow + c * 128 + lane * 4);
#pragma unroll
    for (int j = 0; j < 4; ++j) {
      float rl = fmaxf(a4[j], 0.f);
      unsigned kb = __float_as_uint(rl + d4[j] * 1e-4f); // keys >= 0
      key[c * 4 + j] = kb;
      kmax = kmax < kb ? kb : kmax;
    }
  }
#else
#pragma unroll
  for (int c = 0; c < CHUNKS; ++c) {
    v4f a4 = *(const v4f*)(Arow + c * 128 + lane * 4);
    v4f d4 = *(const v4f*)(Drow + c * 128 + lane * 4);
    *(v4f*)(&sA[w][c * 128 + lane * 4]) = a4; // ds_store_b128
#pragma unroll
    for (int j = 0; j < 4; ++j) {
      float rl = fmaxf(a4[j], 0.f);
      unsigned kb = __float_as_uint(rl + d4[j] * 1e-4f);
      key[c * 4 + j] = kb;
      kmax = kmax < kb ? kb : kmax;
    }
  }
#endif

  // Exact K-th largest key T: largest v with count(key >= v) >= K.
  unsigned hi = wredMaxU(kmax);
  unsigned lo = 0u;
  while (lo < hi) {
    unsigned mid = lo + ((hi - lo + 1u) >> 1);
    int c = 0;
#pragma unroll
    for (int r = 0; r < ELEMS; ++r) c += (key[r] >= mid) ? 1 : 0;
    c = wredAdd(c);
    if (c >= K_) lo = mid;
    else hi = mid - 1u;
  }
  const unsigned T = lo;

  int cge = 0, cgt = 0;
#pragma unroll
  for (int r = 0; r < ELEMS; ++r) {
    cge += (key[r] >= T) ? 1 : 0;
    cgt += (key[r] > T) ? 1 : 0;
  }
  cge = wredAdd(cge);
  cgt = wredAdd(cgt);
  const int extra = K_ - cgt; // always >= 1
  const bool takeAllEq = ((cge - cgt) == extra);

  int* mycols = wcols + row * K_;
  float* myvals = wvals + row * K_;
  float msum = 0.f;
#pragma unroll
  for (int r = 0; r < ELEMS; ++r) {
    const bool sel = takeAllEq ? (key[r] >= T) : (key[r] > T);
    if (sel) {
      const int col = (r >> 2) * 128 + lane * 4 + (r & 3);
      const float v = fmaxf(sA[w][col], 0.f);
      const unsigned s = atomicAdd(&slot[w], 1u); // ds_add_rtn_u32
      mycols[s] = col;
      myvals[s] = v;
      msum += v;
    }
  }
  if (!takeAllEq) {
    // Rare boundary-duplicate path: take `extra` smallest column indices among
    // key == T (matches jax.lax.top_k lower-index-wins tie-breaking).
    int last = -1;
    for (int s2 = 0; s2 < extra; ++s2) {
      int myidx = N_;
#pragma unroll
      for (int r = 0; r < ELEMS; ++r) {
        const int col = (r >> 2) * 128 + lane * 4 + (r & 3);
        if (key[r] == T && col > last && col < myidx) myidx = col;
      }
      const int m = wredMinI(myidx); // owning lane is unique (cols unique)
      if (myidx == m) {
        const float v = fmaxf(sA[w][m], 0.f);
        const unsigned s = atomicAdd(&slot[w], 1u);
        mycols[s] = m;
        myvals[s] = v;
        msum += v;
      }
      last = m;
    }
  }

  const float rsum = wredAddF(msum);
  if (lane == 0) wdinv[row] = rsqrtf(rsum + 1.0f); // +1 from added identity
}

// Pass 2: one wave32 per row scatters the <=21 scaled nonzeros over the zeros
// written by pass 1 (kernel boundary orders the stores).
__global__ __launch_bounds__(256) void gsp_scatter(
    const int* __restrict__ wcols, const float* __restrict__ wvals,
    const float* __restrict__ wdinv, float* __restrict__ out) {
  const int lane = (int)threadIdx.x & 31;
  const int w = (int)threadIdx.x >> 5;
  const int row = (int)blockIdx.x * 8 + w;
  const int b = row >> 11;      // row / N_
  const int i = row & (N_ - 1); // row % N_
  const float di = wdinv[row];
  const size_t rbase = (size_t)row * N_;

  int col = 0;
  float v = 0.f;
  const bool act = (lane < K_);
  if (act) {
    col = wcols[row * K_ + lane];
    v = wvals[row * K_ + lane];
  }
  const bool isDiag = act && (col == i);
  const unsigned long long anyDiag = __ballot(isDiag);
  if (act) {
    const float dj = wdinv[b * N_ + col];
    out[rbase + col] = isDiag ? (di * di * (v + 1.0f)) : (di * v * dj);
  }
  // Diagonal not in the top-K list: identity contribution alone.
  if (anyDiag == 0ull && lane == 0) out[rbase + i] = di * di;
}

extern "C" void kernel_launch(void* const* d_in, const int* in_sizes, int n_in,
                              void* d_out, int out_size, void* d_ws,
                              size_t ws_size, hipStream_t stream) {
  const float* A = (const float*)d_in[0];
  const float* dope = (const float*)d_in[1];
  float* out = (float*)d_out;

  int* wcols = (int*)d_ws;                                       // NROWS*K ints
  float* wvals = (float*)((char*)d_ws + (size_t)NROWS * K_ * 4); // NROWS*K f32
  float* wdinv = (float*)((char*)d_ws + (size_t)NROWS * K_ * 8); // NROWS f32

  gsp_topk_rows<<<NROWS / WAVES, WAVES * 32, 0, stream>>>(A, dope, out, wcols,
                                                          wvals, wdinv);
  gsp_scatter<<<NROWS / 8, 256, 0, stream>>>(wcols, wvals, wdinv, out);

  (void)in_sizes;
  (void)n_in;
  (void)out_size;
  (void)ws_size;
}